// Gcn_60971355734712
// MI455X (gfx1250) — compile-verified
//
#include <hip/hip_runtime.h>
#include <hip/hip_bf16.h>

#define DIM   64
#define NGRPH 128

typedef __attribute__((ext_vector_type(2))) float v2f;
typedef __attribute__((ext_vector_type(8))) float v8f;

// ---------------------------------------------------------------------------
// Zero a buffer (float4 granularity). Used instead of hipMemsetAsync so the
// whole launch sequence is plain kernels on `stream` (graph-capture safe).
// ---------------------------------------------------------------------------
__global__ void zero_f4(float4* __restrict__ p, int n4) {
    int i = blockIdx.x * blockDim.x + threadIdx.x;
    if (i < n4) p[i] = make_float4(0.f, 0.f, 0.f, 0.f);
}

// ---------------------------------------------------------------------------
// Degree: deg[n] = 1 (self loop) + #incoming edges; then dinv = rsqrt(deg).
// ---------------------------------------------------------------------------
__global__ void deg_init(float* __restrict__ deg, int n) {
    int i = blockIdx.x * blockDim.x + threadIdx.x;
    if (i < n) deg[i] = 1.0f;               // self-loop contribution
}
__global__ void deg_accum(float* __restrict__ deg, const int* __restrict__ dst, int e) {
    int i = blockIdx.x * blockDim.x + threadIdx.x;
    if (i < e) unsafeAtomicAdd(&deg[dst[i]], 1.0f);   // global_atomic_add_f32 @ L2
}
__global__ void deg_rsqrt(float* __restrict__ deg, int n) {
    int i = blockIdx.x * blockDim.x + threadIdx.x;
    if (i < n) { float d = deg[i]; deg[i] = d > 0.f ? rsqrtf(d) : 0.f; }
}

// ---------------------------------------------------------------------------
// C[M,64] = A[M,64] @ W[64,64] via V_WMMA_F32_16X16X4_F32 (exact f32 path).
// Block = 128 threads = 4 waves; block covers 16 rows, wave w covers cols
// [16w,16w+16). K=64 consumed as 16 chained WMMAs accumulating in 8 VGPRs.
// ISA VGPR layout (CDNA5 ISA):
//   A 16x4 f32: lanes 0-15 row M=l hold K={k0,k0+1}; lanes 16-31 K={k0+2,k0+3}
//   B  4x16:   mirrored striping across lanes within each VGPR
//   C/D 16x16: VGPR r = row (r + 8*half), col = colBase + (lane&15)
// M must be a multiple of 16 (100000 = 6250*16).
// ---------------------------------------------------------------------------
__global__ void gemm64_wmma(const float* __restrict__ A, const float* __restrict__ W,
                            float* __restrict__ C, int M) {
    const int wave    = threadIdx.x >> 5;        // 0..3 -> column tile
    const int lane    = threadIdx.x & 31;
    const int half    = lane >> 4;               // 0 or 1
    const int l       = lane & 15;
    const int rowBase = blockIdx.x * 16;
    const int colBase = wave * 16;
    if (rowBase >= M) return;

    v8f acc = {};
#pragma unroll
    for (int k0 = 0; k0 < DIM; k0 += 4) {
        v2f a, b;
        const float* ap = A + (size_t)(rowBase + l) * DIM + (k0 + 2 * half);
        a.x = ap[0];                  // K = k0 + 2*half
        a.y = ap[1];                  // K = k0 + 2*half + 1
        const float* bp = W + (size_t)(k0 + 2 * half) * DIM + colBase + l;
        b.x = bp[0];                  // row K = k0 + 2*half
        b.y = bp[DIM];                // row K = k0 + 2*half + 1
        acc = __builtin_amdgcn_wmma_f32_16x16x4_f32(
                  /*neg_a=*/false, a, /*neg_b=*/false, b,
                  /*c_mod=*/(short)0, acc, /*reuse_a=*/false, /*reuse_b=*/false);
    }
#pragma unroll
    for (int r = 0; r < 8; ++r)
        C[(size_t)(rowBase + r + 8 * half) * DIM + colBase + l] = acc[r];
}

// ---------------------------------------------------------------------------
// Edge scatter: one wave32 per edge. Lane j moves float2 j of the 64-float
// row -> one fully coalesced 256B gather of xw[src], then 64 L2-resident
// float atomics into agg[dst]. xw/agg are 25.6MB each => live in 192MB L2.
// ---------------------------------------------------------------------------
__global__ void edge_scatter(const float* __restrict__ xw, float* __restrict__ agg,
                             const float* __restrict__ dinv,
                             const int* __restrict__ src, const int* __restrict__ dst,
                             int E) {
    int wid  = (blockIdx.x * blockDim.x + threadIdx.x) >> 5;
    int lane = threadIdx.x & 31;
    if (wid >= E) return;
    int s = src[wid];
    int d = dst[wid];
    float nrm = dinv[s] * dinv[d];
    float2 v = ((const float2*)(xw + (size_t)s * DIM))[lane];
    float* ap = agg + (size_t)d * DIM + 2 * lane;
    unsafeAtomicAdd(ap,     v.x * nrm);
    unsafeAtomicAdd(ap + 1, v.y * nrm);
}

// ---------------------------------------------------------------------------
// Epilogue (fused): agg[i] += xw[i]*dinv[n]^2 (self-loop) + bias; optional ReLU.
// In-place on agg, which becomes the next layer's h.
// ---------------------------------------------------------------------------
__global__ void finalize(const float* __restrict__ xw, float* __restrict__ agg,
                         const float* __restrict__ dinv, const float* __restrict__ bias,
                         int n64, int doRelu) {
    int i = blockIdx.x * blockDim.x + threadIdx.x;
    if (i >= n64) return;
    int n = i >> 6, d = i & 63;
    float di = dinv[n];
    float v = agg[i] + xw[i] * di * di + bias[d];
    if (doRelu) v = fmaxf(v, 0.f);
    agg[i] = v;
}

// ---------------------------------------------------------------------------
// Global mean pool + final linear.
// ---------------------------------------------------------------------------
__global__ void pool_accum(const float* __restrict__ h, const int* __restrict__ batch,
                           float* __restrict__ sums, int n64) {
    int i = blockIdx.x * blockDim.x + threadIdx.x;
    if (i >= n64) return;
    int n = i >> 6, d = i & 63;
    unsafeAtomicAdd(&sums[(size_t)batch[n] * DIM + d], h[i]);
}
__global__ void pool_count(const int* __restrict__ batch, float* __restrict__ cnt, int n) {
    int i = blockIdx.x * blockDim.x + threadIdx.x;
    if (i < n) unsafeAtomicAdd(&cnt[batch[i]], 1.0f);
}
__global__ void pool_linear(const float* __restrict__ sums, const float* __restrict__ cnt,
                            const float* __restrict__ Wlin, const float* __restrict__ blin,
                            float* __restrict__ out) {
    int g = blockIdx.x;          // 128
    int j = threadIdx.x;         // 64
    float inv = 1.0f / fmaxf(cnt[g], 1.0f);
    float acc = blin[j];
#pragma unroll
    for (int k = 0; k < DIM; ++k)
        acc += (sums[g * DIM + k] * inv) * Wlin[k * DIM + j];
    out[g * DIM + j] = acc;
}

// ---------------------------------------------------------------------------
// One GCN layer: zero agg, WMMA GEMM, edge scatter, fused epilogue.
// ---------------------------------------------------------------------------
static void run_layer(const float* h_in, const float* W, const float* b,
                      float* XW, float* AGG, const float* dinv,
                      const int* src, const int* dst,
                      int N, int E, int doRelu, hipStream_t stream) {
    const int n64 = N * DIM;
    zero_f4<<<(n64 / 4 + 255) / 256, 256, 0, stream>>>((float4*)AGG, n64 / 4);
    gemm64_wmma<<<(N + 15) / 16, 128, 0, stream>>>(h_in, W, XW, N);
    edge_scatter<<<(E + 7) / 8, 256, 0, stream>>>(XW, AGG, dinv, src, dst, E);
    finalize<<<(n64 + 255) / 256, 256, 0, stream>>>(XW, AGG, dinv, b, n64, doRelu);
}

extern "C" void kernel_launch(void* const* d_in, const int* in_sizes, int n_in,
                              void* d_out, int out_size, void* d_ws, size_t ws_size,
                              hipStream_t stream) {
    const float* x     = (const float*)d_in[0];
    const int*   ei    = (const int*)  d_in[1];
    const int*   batch = (const int*)  d_in[2];
    const float* W1 = (const float*)d_in[3];  const float* b1 = (const float*)d_in[4];
    const float* W2 = (const float*)d_in[5];  const float* b2 = (const float*)d_in[6];
    const float* W3 = (const float*)d_in[7];  const float* b3 = (const float*)d_in[8];
    const float* Wl = (const float*)d_in[9];  const float* bl = (const float*)d_in[10];
    float* out = (float*)d_out;

    const int N = in_sizes[0] / DIM;      // 100000 (multiple of 16)
    const int E = in_sizes[1] / 2;        // 1,600,000
    const int* src = ei;
    const int* dst = ei + E;

    // Workspace layout (floats); all offsets are 16B-aligned.
    float* ws   = (float*)d_ws;
    float* dinv = ws;                              // N
    float* XW   = dinv + N;                        // N*64
    float* P    = XW + (size_t)N * DIM;            // N*64
    float* Q    = P  + (size_t)N * DIM;            // N*64
    float* sums = Q  + (size_t)N * DIM;            // 128*64
    float* cnt  = sums + NGRPH * DIM;              // 128
    (void)ws_size; (void)n_in; (void)out_size;

    // --- Degrees / symmetric normalization (shared by all 3 layers) ---
    deg_init <<<(N + 255) / 256, 256, 0, stream>>>(dinv, N);
    deg_accum<<<(E + 255) / 256, 256, 0, stream>>>(dinv, dst, E);
    deg_rsqrt<<<(N + 255) / 256, 256, 0, stream>>>(dinv, N);

    // --- 3 GCN layers, ping-ponging the node buffers ---
    run_layer(x, W1, b1, XW, P, dinv, src, dst, N, E, /*relu=*/1, stream);  // H1 = P
    run_layer(P, W2, b2, XW, Q, dinv, src, dst, N, E, /*relu=*/1, stream);  // H2 = Q
    run_layer(Q, W3, b3, XW, P, dinv, src, dst, N, E, /*relu=*/0, stream);  // H3 = P

    // --- Global mean pool + final linear ---
    const int poolF = NGRPH * DIM + NGRPH;   // sums + cnt are contiguous
    zero_f4<<<(poolF / 4 + 255) / 256, 256, 0, stream>>>((float4*)sums, poolF / 4);
    pool_accum<<<(N * DIM + 255) / 256, 256, 0, stream>>>(P, batch, sums, N * DIM);
    pool_count<<<(N + 255) / 256, 256, 0, stream>>>(batch, cnt, N);
    pool_linear<<<NGRPH, DIM, 0, stream>>>(sums, cnt, Wl, bl, out);
}